// GraphSAINT_73735998538337
// MI455X (gfx1250) — compile-verified
//
#include <hip/hip_runtime.h>
#include <hip/hip_bf16.h>

typedef __attribute__((ext_vector_type(2))) float v2f;
typedef __attribute__((ext_vector_type(4))) float v4f;
typedef __attribute__((ext_vector_type(8))) float v8f;

// ---------------------------------------------------------------------------
// GEMM: Y[N, NT*16] = X[N,128] @ W[cout_real,128]^T  (+ bias, zero-padded)
// fp32 WMMA 16x16x4. One 256-thread block = 8 waves, each wave owns one
// 16-row M-tile and iterates all NT N-tiles.
//
// W is staged in LDS pre-swizzled into *fragment order*: fragment (k0idx, nt)
// occupies 32 consecutive float2 slots indexed by lane, so every DS read in
// the K-loop is 32 lanes x 8B contiguous (256B) -> conflict-free across all
// 64 LDS banks (vs 16-way conflicts with a row-major [COUT][128] layout).
// ---------------------------------------------------------------------------
template <int NT, bool HAS_BIAS>
__global__ __launch_bounds__(256) void gemm_wmma(
    const float* __restrict__ X, const float* __restrict__ W,
    const float* __restrict__ bias, float* __restrict__ Y,
    int nrows, int cout_real)
{
    constexpr int COUT  = NT * 16;
    constexpr int NFRAG = 32 * NT;          // (k0idx, nt) fragment count
    __shared__ v2f bfrag[NFRAG * 32];       // 64 KB for NT=8, 8 KB for NT=1

    const int tid = threadIdx.x;

    // Stage W into LDS in B-fragment order (zero-pad rows >= cout_real).
    // Consecutive threads write consecutive float2 slots -> conflict-free.
    for (int f = tid; f < NFRAG * 32; f += 256) {
        const int lane  = f & 31;
        const int nt    = (f >> 5) % NT;
        const int k0idx = f / (32 * NT);
        const int lmod  = lane & 15;
        const int lhalf = lane >> 4;
        const int n = nt * 16 + lmod;
        const int k = k0idx * 4 + 2 * lhalf;
        v2f v = {0.f, 0.f};
        if (n < cout_real) v = *(const v2f*)(W + n * 128 + k);
        bfrag[f] = v;
    }
    __syncthreads();

    const int wave  = tid >> 5;
    const int lane  = tid & 31;
    const int lhalf = lane >> 4;   // 0: K pair {0,1} / rows M+0, 1: {2,3} / M+8
    const int lmod  = lane & 15;

    const long r0 = (long)blockIdx.x * 128 + wave * 16;
    long arow = r0 + lmod;
    if (arow >= nrows) arow = nrows - 1;          // clamp; store is masked
    const float* aptr = X + arow * 128 + 2 * lhalf;
    const v2f* bbase = bfrag + lane;

    v8f acc[NT];
#pragma unroll
    for (int nt = 0; nt < NT; ++nt) {
        v8f z = {0.f, 0.f, 0.f, 0.f, 0.f, 0.f, 0.f, 0.f};
        acc[nt] = z;
    }

    for (int k0idx = 0; k0idx < 32; ++k0idx) {
        // A 16x4 fp32 fragment: lane holds X[m][4*k0idx + 2*lhalf .. +1]
        v2f a = *(const v2f*)(aptr + k0idx * 4);
#pragma unroll
        for (int nt = 0; nt < NT; ++nt) {
            // B 4x16 fp32 fragment, swizzled: 32 contiguous float2 per fragment
            v2f b = bbase[(k0idx * NT + nt) * 32];
            acc[nt] = __builtin_amdgcn_wmma_f32_16x16x4_f32(
                false, a, false, b, (short)0, acc[nt], false, false);
        }
    }

    // D layout: VGPR v -> row r0 + v + 8*lhalf, col nt*16 + lmod
#pragma unroll
    for (int nt = 0; nt < NT; ++nt) {
        const int n = nt * 16 + lmod;
        float badd = 0.f;
        if (HAS_BIAS && n < cout_real) badd = bias[n];
#pragma unroll
        for (int v = 0; v < 8; ++v) {
            long m = r0 + v + 8 * lhalf;
            if (m < nrows) {
                float val = acc[nt][v];
                if (HAS_BIAS) val += badd;
                Y[m * COUT + n] = val;
            }
        }
    }
}

// ---------------------------------------------------------------------------
// Edge scatter: acc[dst] += yrel[src] * ew   (C = CVEC*4 channels, fp32 atomics)
// ---------------------------------------------------------------------------
template <int CVEC>
__global__ __launch_bounds__(256) void scatter_add(
    const float* __restrict__ yrel, const float* __restrict__ ew,
    const int* __restrict__ src, const int* __restrict__ dst,
    float* __restrict__ acc, int E)
{
    const long t = (long)blockIdx.x * 256 + threadIdx.x;
    const long e = t / CVEC;
    const int  g = (int)(t % CVEC);
    if (e >= E) return;
    const int   s = src[e];
    const int   d = dst[e];
    const float w = ew[e];
    constexpr int C = CVEC * 4;
    v4f m = *(const v4f*)(yrel + (long)s * C + g * 4);
    float* o = acc + (long)d * C + g * 4;
    atomicAdd(o + 0, m.x * w);
    atomicAdd(o + 1, m.y * w);
    atomicAdd(o + 2, m.z * w);
    atomicAdd(o + 3, m.w * w);
}

// ---------------------------------------------------------------------------
// BatchNorm statistics: stats[c] = sum, stats[128+c] = sum of squares
// ---------------------------------------------------------------------------
__global__ __launch_bounds__(256) void zero256(float* __restrict__ p)
{
    p[threadIdx.x] = 0.f;
}

__global__ __launch_bounds__(256) void bn_stats(
    const float* __restrict__ X, float* __restrict__ stats, int nrows)
{
    __shared__ float ssum[256];
    __shared__ float ssq[256];
    const int tid  = threadIdx.x;
    const int c    = tid & 127;
    const int half = tid >> 7;
    const long r0  = (long)blockIdx.x * 256;
    long rend = r0 + 256;
    if (rend > nrows) rend = nrows;
    float s = 0.f, q = 0.f;
    for (long r = r0 + half; r < rend; r += 2) {
        float v = X[r * 128 + c];
        s += v;
        q += v * v;
    }
    ssum[tid] = s;
    ssq[tid]  = q;
    __syncthreads();
    if (half == 0) {
        atomicAdd(&stats[c],       ssum[tid] + ssum[tid + 128]);
        atomicAdd(&stats[128 + c], ssq[tid]  + ssq[tid + 128]);
    }
}

__global__ __launch_bounds__(256) void bn_apply_relu(
    const float* __restrict__ X, const float* __restrict__ stats,
    const float* __restrict__ gamma, const float* __restrict__ beta,
    float* __restrict__ Y, long total, float invN)
{
    const long i = (long)blockIdx.x * 256 + threadIdx.x;
    if (i >= total) return;
    const int c = (int)(i & 127);
    const float mean = stats[c] * invN;
    const float var  = stats[128 + c] * invN - mean * mean;   // biased variance
    const float rstd = __frsqrt_rn(var + 1e-5f);
    float v = (X[i] - mean) * rstd * gamma[c] + beta[c];
    Y[i] = v > 0.f ? v : 0.f;
}

// ---------------------------------------------------------------------------
// log_softmax over first 9 of 16 padded columns
// ---------------------------------------------------------------------------
__global__ __launch_bounds__(256) void logsoftmax9(
    const float* __restrict__ X16, float* __restrict__ out, int nrows)
{
    const long r = (long)blockIdx.x * 256 + threadIdx.x;
    if (r >= nrows) return;
    float v[9];
    float m = -1e30f;
#pragma unroll
    for (int j = 0; j < 9; ++j) {
        v[j] = X16[r * 16 + j];
        m = fmaxf(m, v[j]);
    }
    float sum = 0.f;
#pragma unroll
    for (int j = 0; j < 9; ++j) sum += __expf(v[j] - m);
    const float lse = m + __logf(sum);
#pragma unroll
    for (int j = 0; j < 9; ++j) out[r * 9 + j] = v[j] - lse;
}

// ---------------------------------------------------------------------------
extern "C" void kernel_launch(void* const* d_in, const int* in_sizes, int n_in,
                              void* d_out, int out_size, void* d_ws, size_t ws_size,
                              hipStream_t stream)
{
    const float* x      = (const float*)d_in[0];
    const int*   ei     = (const int*)d_in[1];
    const float* ew     = (const float*)d_in[2];
    const float* Wrel0  = (const float*)d_in[3];
    const float* brel0  = (const float*)d_in[4];
    const float* Wroot0 = (const float*)d_in[5];
    const float* gamma0 = (const float*)d_in[6];
    const float* beta0  = (const float*)d_in[7];
    const float* Wrel1  = (const float*)d_in[8];
    const float* brel1  = (const float*)d_in[9];
    const float* Wroot1 = (const float*)d_in[10];
    const float* gamma1 = (const float*)d_in[11];
    const float* beta1  = (const float*)d_in[12];
    const float* Wrel2  = (const float*)d_in[13];
    const float* brel2  = (const float*)d_in[14];
    const float* Wroot2 = (const float*)d_in[15];
    float* out = (float*)d_out;

    const int N = in_sizes[0] / 128;
    const int E = in_sizes[2];
    const int* src = ei;
    const int* dst = ei + E;

    float* ws    = (float*)d_ws;
    float* h     = ws;                         // [N,128]
    float* yrel  = ws + (size_t)N * 128;       // [N,128] (layer2: [N,16])
    float* acc   = ws + (size_t)N * 128 * 2;   // [N,128] (layer2: [N,16])
    float* stats = ws + (size_t)N * 128 * 3;   // [256]

    const dim3 b256(256);
    const int gemm_grid = (N + 127) / 128;
    const int sg128     = (int)(((long)E * 32 + 255) / 256);
    const int sg16      = (int)(((long)E * 4 + 255) / 256);
    const int ew_grid   = (int)(((long)N * 128 + 255) / 256);
    const int row_grid  = (N + 255) / 256;
    const float invN    = 1.0f / (float)N;

    // ---- layer 0: conv(x) -> BN -> ReLU -> h ----
    gemm_wmma<8, false><<<gemm_grid, b256, 0, stream>>>(x, Wrel0, nullptr, yrel, N, 128);
    gemm_wmma<8, true ><<<gemm_grid, b256, 0, stream>>>(x, Wroot0, brel0, acc, N, 128);
    scatter_add<32><<<sg128, b256, 0, stream>>>(yrel, ew, src, dst, acc, E);
    zero256<<<1, b256, 0, stream>>>(stats);
    bn_stats<<<row_grid, b256, 0, stream>>>(acc, stats, N);
    bn_apply_relu<<<ew_grid, b256, 0, stream>>>(acc, stats, gamma0, beta0, h,
                                                (long)N * 128, invN);

    // ---- layer 1: conv(h) -> BN -> ReLU -> h ----
    gemm_wmma<8, false><<<gemm_grid, b256, 0, stream>>>(h, Wrel1, nullptr, yrel, N, 128);
    gemm_wmma<8, true ><<<gemm_grid, b256, 0, stream>>>(h, Wroot1, brel1, acc, N, 128);
    scatter_add<32><<<sg128, b256, 0, stream>>>(yrel, ew, src, dst, acc, E);
    zero256<<<1, b256, 0, stream>>>(stats);
    bn_stats<<<row_grid, b256, 0, stream>>>(acc, stats, N);
    bn_apply_relu<<<ew_grid, b256, 0, stream>>>(acc, stats, gamma1, beta1, h,
                                                (long)N * 128, invN);

    // ---- layer 2: conv(h) (cout=9, padded to 16) -> log_softmax ----
    gemm_wmma<1, false><<<gemm_grid, b256, 0, stream>>>(h, Wrel2, nullptr, yrel, N, 9);
    gemm_wmma<1, true ><<<gemm_grid, b256, 0, stream>>>(h, Wroot2, brel2, acc, N, 9);
    scatter_add<4><<<sg16, b256, 0, stream>>>(yrel, ew, src, dst, acc, E);
    logsoftmax9<<<row_grid, b256, 0, stream>>>(acc, out, N);
}